// NERCombinedModel_81896436400347
// MI455X (gfx1250) — compile-verified
//
#include <hip/hip_runtime.h>
#include <hip/hip_bf16.h>

// ---------------------------------------------------------------------------
// NER combined model on MI455X (gfx1250, wave32).
//   1) Fused gather + input-projection GEMMs with V_WMMA_F32_16X16X4_F32
//      (full fp32 precision):  Xih[t][r] = X[t] . Wih[r] + bih[r] + bhh[r]
//   2) Fused LSTM recurrences: 50 workgroups (25 per LSTM). Each WG owns 20
//      h-lanes (80 gate rows) and keeps its 80x500 f32 Whh slice resident in
//      LDS (156 KB -- possible only with CDNA5's 320 KB LDS/WG). One
//      device-scope atomic barrier per step, ping-pong h buffers in L2.
//   3) Tiny FC: out = [h_doc, h_ner] @ fcW.T + fcb.
// ---------------------------------------------------------------------------

typedef __attribute__((ext_vector_type(2))) float v2f;
typedef __attribute__((ext_vector_type(8))) float v8f;

#define FOURH 2000   // 4*H
#define HDIM  500
#define NTN   125    // 2000/16 output-row tiles
#define NWG   25     // workgroups per LSTM (25 * 20 = 500 h lanes)

// ---------------------------------------------------------------------------
// Kernel 1: gather + GEMM via fp32 WMMA. One wave computes one 16x16 tile of
// Xih, accumulating K/4 WMMA ops.  A-fragment (16x4 f32): lane L holds
// A[L%16][kh + v] with kh = 2*(L/16), v = vgpr 0/1.  B-fragment (4x16 f32):
// lane L holds B[kh + v][L%16].  C/D: VGPR v, lane L -> row v + 8*(L/16),
// col L%16.  All tile counts divide exactly; EXEC stays all-ones (WMMA req).
// ---------------------------------------------------------------------------
__global__ __launch_bounds__(256) void wmma_xih_kernel(
    const float* __restrict__ embA,   // emb table [100000,300]
    const float* __restrict__ embB,   // emb_ner [64,300] (concat mode only)
    const int*   __restrict__ idxA,   // seq / ner_words
    const int*   __restrict__ idxB,   // ner_labels (concat mode only)
    const float* __restrict__ W,      // [2000, K]
    const float* __restrict__ bih,
    const float* __restrict__ bhh,
    float*       __restrict__ Xih,    // [T, 2000]
    int K, int concatMode)
{
    const int lane = threadIdx.x & 31;
    const int wave = threadIdx.x >> 5;
    const int tile = blockIdx.x * 8 + wave;   // tile counts are exact multiples of 8
    const int tM = tile / NTN;
    const int tN = tile % NTN;

    const int m  = lane & 15;
    const int kh = (lane >> 4) << 1;          // 0 or 2
    const int t  = tM * 16 + m;               // time row this lane loads for A
    const int n  = tN * 16 + m;               // W row this lane loads for B

    const float* arow0;
    const float* arow1;
    int split;
    if (!concatMode) {
        arow0 = embA + (long)idxA[t] * 300;
        arow1 = arow0;
        split = K;                            // never switches
    } else {
        arow0 = embA + (long)idxA[t] * 300;
        arow1 = embB + (long)idxB[t] * 300 - 300;  // arow1[kk] == emb_ner[..][kk-300]
        split = 300;
    }
    const float* wrow = W + (long)n * K;

    v8f c = {};
    for (int k0 = 0; k0 < K; k0 += 4) {
        const int kk = k0 + kh;               // even, so (kk,kk+1) never straddles 300
        const float* ap = (kk < split) ? arow0 : arow1;
        v2f a; a.x = ap[kk];   a.y = ap[kk + 1];
        v2f b; b.x = wrow[kk]; b.y = wrow[kk + 1];
        c = __builtin_amdgcn_wmma_f32_16x16x4_f32(
                /*neg_a=*/false, a, /*neg_b=*/false, b,
                /*c_mod=*/(short)0, c, /*reuse_a=*/false, /*reuse_b=*/false);
    }

    const float bias = bih[n] + bhh[n];
    const int rowbase = tM * 16 + ((lane >> 4) << 3);
#pragma unroll
    for (int v = 0; v < 8; ++v) {
        Xih[(long)(rowbase + v) * FOURH + n] = c[v] + bias;
    }
}

// ---------------------------------------------------------------------------
// Kernel 2: fused sequential LSTM recurrences (doc + ner run concurrently).
// Blocks 0..24 -> doc LSTM, 25..49 -> ner LSTM.  160 threads/WG.
// WG w (within its group) owns h-indices [20w, 20w+20) -> 80 gate rows
// r = q*500 + jj for quadrants q=i,f,g,o.  Whh slice lives in LDS as
// whh_s[k][rl] (rl = q*20 + local), loaded once from Whh row-major
// (coalesced reads).  Hot loop: thread t = half*80 + rl accumulates a
// 250-deep partial dot; conflict-free LDS banking (stride 80 words).
// One global barrier per step; h double-buffered (one barrier suffices:
// writes to buf[s&1] at step s+2 are ordered after barrier(s+1), which is
// after every WG finished re-reading buf[s&1] at step s).
// ---------------------------------------------------------------------------
__global__ __launch_bounds__(160) void lstm_rec_kernel(
    const float* __restrict__ Xih_d, const float* __restrict__ Whh_d,
    float* __restrict__ hbuf_d, float* __restrict__ hout_d,
    unsigned* __restrict__ ctr_d, int T_d,
    const float* __restrict__ Xih_n, const float* __restrict__ Whh_n,
    float* __restrict__ hbuf_n, float* __restrict__ hout_n,
    unsigned* __restrict__ ctr_n, int T_n)
{
    __shared__ float whh_s[80 * HDIM];   // 156.25 KB (CDNA5: 320 KB LDS/WG)
    __shared__ float h_s[512];
    __shared__ float part_s[160];
    __shared__ float gates_s[80];

    const int grp = blockIdx.x / NWG;            // 0 = doc, 1 = ner
    const int w   = blockIdx.x % NWG;
    const float* __restrict__ Xih  = grp ? Xih_n  : Xih_d;
    const float* __restrict__ Whh  = grp ? Whh_n  : Whh_d;
    float*    hbuf    = grp ? hbuf_n : hbuf_d;
    float*    hout    = grp ? hout_n : hout_d;
    unsigned* counter = grp ? ctr_n  : ctr_d;
    const int T       = grp ? T_n    : T_d;

    const int tid  = threadIdx.x;
    const int half = tid / 80;                   // 0 or 1
    const int rl   = tid % 80;                   // local gate row
    const int q    = rl / 20;                    // gate quadrant
    const int jj   = w * 20 + (rl % 20);         // owned h index (tid<20)
    const int r    = q * HDIM + jj;              // global gate row

    // one-time weight preload: coalesced global reads, scattered LDS writes
    for (int idx = tid; idx < 80 * HDIM; idx += 160) {
        const int prl = idx / HDIM;
        const int k   = idx % HDIM;
        const int gr  = (prl / 20) * HDIM + w * 20 + (prl % 20);
        whh_s[k * 80 + prl] = Whh[(long)gr * HDIM + k];
    }
    for (int i = tid; i < 512; i += 160) h_s[i] = 0.f;
    __syncthreads();

    float cstate = 0.f;
    const int kbase = half * 250;

    for (int step = 0; step < T; ++step) {
        // 250-deep partial dot (half 0 also folds in the precomputed Xih+bias)
        float acc = (half == 0) ? Xih[(long)step * FOURH + r] : 0.f;
#pragma unroll 5
        for (int k = 0; k < 250; ++k)
            acc += whh_s[(kbase + k) * 80 + rl] * h_s[kbase + k];
        part_s[tid] = acc;
        if (half == 0 && step + 1 < T)   // warm L2/L0 for next step's Xih row
            __builtin_prefetch(&Xih[(long)(step + 1) * FOURH + r], 0, 0);
        __syncthreads();

        if (tid < 80) gates_s[tid] = part_s[tid] + part_s[tid + 80];
        __syncthreads();

        if (tid < 20) {
            const float gi = gates_s[tid];
            const float gf = gates_s[tid + 20];
            const float gg = gates_s[tid + 40];
            const float go = gates_s[tid + 60];
            const float i_ = 1.f / (1.f + __expf(-gi));
            const float f_ = 1.f / (1.f + __expf(-gf));
            const float g_ = tanhf(gg);
            const float o_ = 1.f / (1.f + __expf(-go));
            cstate = f_ * cstate + i_ * g_;
            const float h = o_ * tanhf(cstate);
            hbuf[(step & 1) * 512 + jj] = h;
            if (step == T - 1) hout[jj] = h;
        }
        __threadfence();                 // publish h stores at device scope
        __syncthreads();

        if (tid == 0) {
            __hip_atomic_fetch_add(counter, 1u, __ATOMIC_RELEASE,
                                   __HIP_MEMORY_SCOPE_AGENT);
            const unsigned target = (unsigned)NWG * (unsigned)(step + 1);
            while (__hip_atomic_load(counter, __ATOMIC_RELAXED,
                                     __HIP_MEMORY_SCOPE_AGENT) < target)
                __builtin_amdgcn_s_sleep(1);
        }
        __syncthreads();
        __threadfence();                 // acquire other WGs' h stores

        for (int i = tid; i < HDIM; i += 160)
            h_s[i] = hbuf[(step & 1) * 512 + i];
        __syncthreads();
    }
}

// ---------------------------------------------------------------------------
// Kernel 3: final FC, out[20] = fcb + fcW @ [h_doc ; h_ner]
// ---------------------------------------------------------------------------
__global__ __launch_bounds__(64) void fc_kernel(
    const float* __restrict__ hcat,   // [1000]
    const float* __restrict__ fcW,    // [20, 1000]
    const float* __restrict__ fcb,
    float*       __restrict__ out)
{
    const int o = threadIdx.x;
    if (o < 20) {
        float acc = fcb[o];
        const float* wr = fcW + o * 1000;
        for (int i = 0; i < 1000; ++i) acc += wr[i] * hcat[i];
        out[o] = acc;
    }
}

__global__ void init_kernel(unsigned* counters)
{
    if (threadIdx.x < 2) counters[threadIdx.x] = 0u;
}

// ---------------------------------------------------------------------------
extern "C" void kernel_launch(void* const* d_in, const int* in_sizes, int n_in,
                              void* d_out, int out_size, void* d_ws, size_t ws_size,
                              hipStream_t stream)
{
    (void)n_in; (void)out_size; (void)ws_size;

    const int*   seq        = (const int*)d_in[0];
    const int*   ner_words  = (const int*)d_in[1];
    const int*   ner_labels = (const int*)d_in[2];
    const float* emb        = (const float*)d_in[3];
    const float* emb_ner    = (const float*)d_in[4];
    const float* Wih        = (const float*)d_in[5];
    const float* Whh        = (const float*)d_in[6];
    const float* bih        = (const float*)d_in[7];
    const float* bhh        = (const float*)d_in[8];
    const float* Wih_n      = (const float*)d_in[9];
    const float* Whh_n      = (const float*)d_in[10];
    const float* bih_n      = (const float*)d_in[11];
    const float* bhh_n      = (const float*)d_in[12];
    const float* fcW        = (const float*)d_in[13];
    const float* fcb        = (const float*)d_in[14];
    float* out = (float*)d_out;

    const int Td = in_sizes[0];   // 4096
    const int Tn = in_sizes[1];   // 1024

    // --- workspace layout (~41 MB) ---
    char* ws = (char*)d_ws;
    size_t off = 0;
    float* Xih_doc = (float*)(ws + off); off += (size_t)Td * FOURH * 4;
    float* Xih_ner = (float*)(ws + off); off += (size_t)Tn * FOURH * 4;
    float* hbuf_d  = (float*)(ws + off); off += 2 * 512 * 4;
    float* hbuf_n  = (float*)(ws + off); off += 2 * 512 * 4;
    float* hfinal  = (float*)(ws + off); off += 1024 * 4;
    unsigned* ctrs = (unsigned*)(ws + off); off += 256;

    // reset barrier counters every call (graph-replay safe)
    init_kernel<<<1, 32, 0, stream>>>(ctrs);

    // input projections (fused gather + fp32 WMMA GEMM + bias)
    const int blocksDoc = (Td / 16) * NTN / 8;   // 4000
    const int blocksNer = (Tn / 16) * NTN / 8;   // 1000
    wmma_xih_kernel<<<blocksDoc, 256, 0, stream>>>(
        emb, emb, seq, seq, Wih, bih, bhh, Xih_doc, 300, 0);
    wmma_xih_kernel<<<blocksNer, 256, 0, stream>>>(
        emb, emb_ner, ner_words, ner_labels, Wih_n, bih_n, bhh_n, Xih_ner, 600, 1);

    // both sequential recurrences fused into one concurrent launch
    lstm_rec_kernel<<<2 * NWG, 160, 0, stream>>>(
        Xih_doc, Whh,   hbuf_d, hfinal,       ctrs + 0, Td,
        Xih_ner, Whh_n, hbuf_n, hfinal + 500, ctrs + 1, Tn);

    // final classifier
    fc_kernel<<<1, 64, 0, stream>>>(hfinal, fcW, fcb, out);
}